// FastEdgeToEdgeAttention_27041114095820
// MI455X (gfx1250) — compile-verified
//
#include <hip/hip_runtime.h>
#include <math.h>

#define E_TOTAL 200000
#define C       256
#define H       8
#define HD      32
#define EB      64                 // edges per workgroup tile
#define NB      (E_TOTAL / EB)     // 3125 (exact, no tail)
#define THREADS 256

typedef __attribute__((ext_vector_type(2))) float v2f;
typedef __attribute__((ext_vector_type(8))) float v8f;

// f32 WMMA: D(16x16) = A(16x4) * B(4x16) + C, exact fp32 math (matches reference).
__device__ __forceinline__ v8f wmma4(v2f a, v2f b, v8f c) {
  return __builtin_amdgcn_wmma_f32_16x16x4_f32(
      /*neg_a=*/false, a, /*neg_b=*/false, b,
      /*c_mod=*/(short)0, c, /*reuse_a=*/false, /*reuse_b=*/false);
}

// 64x16 output tile, K=256. A rows come from LDS (row-major 64x256 tile),
// B column-block comes from 16 rows of W (out = x @ W^T, so B[k][n] = W[j0+n][k]).
// Per-lane: row = lane%16 (both the A-row-within-tile and the W row j0+row),
// koff = (lane/16)*2 selects the K half per ISA A/B layout.
__device__ __forceinline__ void gemm64x16_K256(const float* __restrict__ a_lds,
                                               const float* __restrict__ w_row,
                                               int row, int koff,
                                               v8f& r0, v8f& r1, v8f& r2, v8f& r3) {
  const v2f* __restrict__ wr = (const v2f*)w_row;
  const v2f* __restrict__ x0 = (const v2f*)(a_lds + (0 * 16 + row) * C);
  const v2f* __restrict__ x1 = (const v2f*)(a_lds + (1 * 16 + row) * C);
  const v2f* __restrict__ x2 = (const v2f*)(a_lds + (2 * 16 + row) * C);
  const v2f* __restrict__ x3 = (const v2f*)(a_lds + (3 * 16 + row) * C);
#pragma unroll 8
  for (int c0 = 0; c0 < C; c0 += 4) {
    const int ci = (c0 + koff) >> 1;    // index in 8-byte units
    const v2f b = wr[ci];
    r0 = wmma4(x0[ci], b, r0);
    r1 = wmma4(x1[ci], b, r1);
    r2 = wmma4(x2[ci], b, r2);
    r3 = wmma4(x3[ci], b, r3);
  }
}

// Write one 16x16 accumulator tile to an LDS 64x256 buffer.
// C/D layout: element vv of v8f is matrix row (vv + mofs), lane%16 is column.
__device__ __forceinline__ void emit_kv(float* __restrict__ dst, v8f acc, int et,
                                        int mofs, int col, float bj, int do_elu) {
#pragma unroll
  for (int vv = 0; vv < 8; ++vv) {
    float val = acc[vv] + bj;
    if (do_elu) val = (val > 0.0f) ? (val + 1.0f) : __expf(val);  // elu(x)+1
    dst[(et * 16 + vv + mofs) * C + col] = val;
  }
}

__device__ __forceinline__ void emit_s(float* __restrict__ ss,
                                       const float* __restrict__ zs, v8f acc,
                                       int et, int mofs, int h, int col) {
#pragma unroll
  for (int vv = 0; vv < 8; ++vv) {
    const int er = et * 16 + vv + mofs;
    ss[er * C + col] = acc[vv] / zs[er * H + h];  // zs already includes +1e-6
  }
}

__device__ __forceinline__ void emit_y(float* __restrict__ ys,
                                       const float* __restrict__ xs, v8f acc,
                                       int et, int mofs, int col, float bj) {
#pragma unroll
  for (int vv = 0; vv < 8; ++vv) {
    const int er = et * 16 + vv + mofs;
    const float val = acc[vv] + bj;
    const float g = 0.5f * val * (1.0f + erff(val * 0.70710678118654752f));
    ys[er * C + col] = xs[er * C + col] + g;  // residual + exact GELU
  }
}

__global__ void zero_ws_kernel(float* __restrict__ ws, int n) {
  const int i = blockIdx.x * blockDim.x + threadIdx.x;
  if (i < n) ws[i] = 0.0f;
}

// ---------------- Pass 1: k,v projections + global kv / k_sum accumulation ----
__global__ void __launch_bounds__(THREADS) pass1_kv(
    const float* __restrict__ x, const float* __restrict__ Wk,
    const float* __restrict__ bk, const float* __restrict__ Wv,
    const float* __restrict__ bv, float* __restrict__ kv_acc,
    float* __restrict__ ksum_acc) {
  extern __shared__ float lds[];
  float* xs = lds;               // 64*256
  float* ks = lds + EB * C;      // 64*256
  float* vs = lds + 2 * EB * C;  // 64*256

  const int tid = threadIdx.x;
  const int lane = tid & 31;
  const int wave = tid >> 5;
  const long e0 = (long)blockIdx.x * EB;

  {  // stage x tile, 16B-vectorized, fully coalesced
    const float4* __restrict__ xg = (const float4*)(x + e0 * C);
    float4* __restrict__ xl = (float4*)xs;
    for (int i = tid; i < EB * C / 4; i += THREADS) xl[i] = xg[i];
  }
  __syncthreads();

  const int row = lane & 15;
  const int koff = (lane >> 4) << 1;
  const int mofs = (lane >> 4) << 3;

  for (int mat = 0; mat < 2; ++mat) {
    const float* __restrict__ W = (mat == 0) ? Wk : Wv;
    const float* __restrict__ bias = (mat == 0) ? bk : bv;
    float* __restrict__ dst = (mat == 0) ? ks : vs;
    for (int jt = 0; jt < 2; ++jt) {
      const int j0 = wave * 32 + jt * 16;
      v8f r0 = {}, r1 = {}, r2 = {}, r3 = {};
      gemm64x16_K256(xs, W + (size_t)(j0 + row) * C, row, koff, r0, r1, r2, r3);
      const float bj = bias[j0 + row];
      emit_kv(dst, r0, 0, mofs, j0 + row, bj, mat == 0);
      emit_kv(dst, r1, 1, mofs, j0 + row, bj, mat == 0);
      emit_kv(dst, r2, 2, mofs, j0 + row, bj, mat == 0);
      emit_kv(dst, r3, 3, mofs, j0 + row, bj, mat == 0);
    }
  }
  __syncthreads();

  // per-head kv += k_tile^T @ v_tile  (tiny: 8*32*32); wave==head, lane==d
  {
    const int h = wave;
    const int d = lane;
    float kvp[HD];
#pragma unroll
    for (int m = 0; m < HD; ++m) kvp[m] = 0.0f;
    float ksp = 0.0f;
    for (int e = 0; e < EB; ++e) {
      const float kk = ks[e * C + h * HD + d];
      ksp += kk;
      const float* __restrict__ vrow = vs + e * C + h * HD;
#pragma unroll
      for (int m = 0; m < HD; ++m) kvp[m] = fmaf(kk, vrow[m], kvp[m]);
    }
    float* __restrict__ kvh = kv_acc + (size_t)(h * HD + d) * HD;
#pragma unroll
    for (int m = 0; m < HD; ++m) atomicAdd(&kvh[m], kvp[m]);
    atomicAdd(&ksum_acc[h * HD + d], ksp);
  }
}

// ---------------- Pass 2: q, num=q@kv, s=num/z, out=s@Wo^T, GELU+res+LN -------
__global__ void __launch_bounds__(THREADS) pass2_out(
    const float* __restrict__ x, const float* __restrict__ Wq,
    const float* __restrict__ bq, const float* __restrict__ kv_acc,
    const float* __restrict__ ksum_acc, const float* __restrict__ Wo,
    const float* __restrict__ bo, const float* __restrict__ gamma,
    const float* __restrict__ beta, float* __restrict__ out) {
  extern __shared__ float lds[];
  float* xs = lds;                   // 16384
  float* qs = xs + EB * C;           // 16384 (reused as ys after num GEMM)
  float* ss = qs + EB * C;           // 16384
  float* kvs = ss + EB * C;          // 8192
  float* ksums = kvs + H * HD * HD;  // 256
  float* zs = ksums + C;             // 512
  float* redS = zs + EB * H;         // 256
  float* redQ = redS + 4 * EB;       // 256
  float* muv = redQ + 4 * EB;        // 64
  float* rsv = muv + EB;             // 64

  const int tid = threadIdx.x;
  const int lane = tid & 31;
  const int wave = tid >> 5;
  const long e0 = (long)blockIdx.x * EB;

  {
    const float4* __restrict__ xg = (const float4*)(x + e0 * C);
    float4* __restrict__ xl = (float4*)xs;
    for (int i = tid; i < EB * C / 4; i += THREADS) xl[i] = xg[i];
    for (int i = tid; i < H * HD * HD; i += THREADS) kvs[i] = kv_acc[i];
    ksums[tid] = ksum_acc[tid];  // tid < 256 == C
  }
  __syncthreads();

  const int row = lane & 15;
  const int koff = (lane >> 4) << 1;
  const int mofs = (lane >> 4) << 3;

  // q = elu(x @ Wq^T + bq) + 1  -> qs
  for (int jt = 0; jt < 2; ++jt) {
    const int j0 = wave * 32 + jt * 16;
    v8f r0 = {}, r1 = {}, r2 = {}, r3 = {};
    gemm64x16_K256(xs, Wq + (size_t)(j0 + row) * C, row, koff, r0, r1, r2, r3);
    const float bj = bq[j0 + row];
    emit_kv(qs, r0, 0, mofs, j0 + row, bj, 1);
    emit_kv(qs, r1, 1, mofs, j0 + row, bj, 1);
    emit_kv(qs, r2, 2, mofs, j0 + row, bj, 1);
    emit_kv(qs, r3, 3, mofs, j0 + row, bj, 1);
  }
  __syncthreads();

  // z[e][h] = q[e,h,:] . k_sum[h,:] + 1e-6
  for (int idx = tid; idx < EB * H; idx += THREADS) {
    const int e = idx >> 3, h = idx & 7;
    const float* __restrict__ qrow = qs + e * C + h * HD;
    const float* __restrict__ kr = ksums + h * HD;
    float z = 1e-6f;
#pragma unroll
    for (int d = 0; d < HD; ++d) z = fmaf(qrow[d], kr[d], z);
    zs[idx] = z;
  }
  __syncthreads();

  // num = q @ kv[h] (per head, K=32, B from LDS), then s = num / z -> ss
  {
    const int h = wave;  // 8 waves == 8 heads
    const float* __restrict__ kvh = kvs + h * HD * HD;
    const v2f* __restrict__ q0 = (const v2f*)(qs + (0 * 16 + row) * C + h * HD);
    const v2f* __restrict__ q1 = (const v2f*)(qs + (1 * 16 + row) * C + h * HD);
    const v2f* __restrict__ q2 = (const v2f*)(qs + (2 * 16 + row) * C + h * HD);
    const v2f* __restrict__ q3 = (const v2f*)(qs + (3 * 16 + row) * C + h * HD);
    for (int nt = 0; nt < 2; ++nt) {
      const int n0 = nt * 16;
      v8f r0 = {}, r1 = {}, r2 = {}, r3 = {};
#pragma unroll
      for (int c0 = 0; c0 < HD; c0 += 4) {
        const int ci = (c0 + koff) >> 1;
        v2f b;
        b.x = kvh[(c0 + koff + 0) * HD + n0 + row];
        b.y = kvh[(c0 + koff + 1) * HD + n0 + row];
        r0 = wmma4(q0[ci], b, r0);
        r1 = wmma4(q1[ci], b, r1);
        r2 = wmma4(q2[ci], b, r2);
        r3 = wmma4(q3[ci], b, r3);
      }
      emit_s(ss, zs, r0, 0, mofs, h, h * HD + n0 + row);
      emit_s(ss, zs, r1, 1, mofs, h, h * HD + n0 + row);
      emit_s(ss, zs, r2, 2, mofs, h, h * HD + n0 + row);
      emit_s(ss, zs, r3, 3, mofs, h, h * HD + n0 + row);
    }
  }
  __syncthreads();

  // out = s @ Wo^T + bo; y = x + gelu(out) -> ys (reuse qs)
  float* __restrict__ ys = qs;
  for (int jt = 0; jt < 2; ++jt) {
    const int j0 = wave * 32 + jt * 16;
    v8f r0 = {}, r1 = {}, r2 = {}, r3 = {};
    gemm64x16_K256(ss, Wo + (size_t)(j0 + row) * C, row, koff, r0, r1, r2, r3);
    const float bj = bo[j0 + row];
    emit_y(ys, xs, r0, 0, mofs, j0 + row, bj);
    emit_y(ys, xs, r1, 1, mofs, j0 + row, bj);
    emit_y(ys, xs, r2, 2, mofs, j0 + row, bj);
    emit_y(ys, xs, r3, 3, mofs, j0 + row, bj);
  }
  __syncthreads();

  // LayerNorm: per-row mean/var (4 column-segments per row, rotated reads)
  {
    const int r = tid & 63;
    const int seg = tid >> 6;
    const float* __restrict__ yr = ys + r * C + seg * 64;
    float s = 0.0f, q2sum = 0.0f;
    for (int i = 0; i < 64; ++i) {
      const int c = (i + lane) & 63;  // rotate to avoid LDS bank conflicts
      const float yv = yr[c];
      s += yv;
      q2sum = fmaf(yv, yv, q2sum);
    }
    redS[seg * EB + r] = s;
    redQ[seg * EB + r] = q2sum;
  }
  __syncthreads();
  if (tid < EB) {
    const float s = redS[tid] + redS[EB + tid] + redS[2 * EB + tid] + redS[3 * EB + tid];
    const float q2sum = redQ[tid] + redQ[EB + tid] + redQ[2 * EB + tid] + redQ[3 * EB + tid];
    const float mu = s * (1.0f / C);
    const float var = q2sum * (1.0f / C) - mu * mu;
    muv[tid] = mu;
    rsv[tid] = rsqrtf(var + 1e-5f);
  }
  __syncthreads();
  {
    float* __restrict__ og = out + e0 * C;
    for (int idx = tid; idx < EB * C; idx += THREADS) {
      const int r = idx >> 8;
      const int c = idx & 255;
      og[idx] = (ys[idx] - muv[r]) * rsv[r] * gamma[c] + beta[c];
    }
  }
}

extern "C" void kernel_launch(void* const* d_in, const int* in_sizes, int n_in,
                              void* d_out, int out_size, void* d_ws, size_t ws_size,
                              hipStream_t stream) {
  (void)in_sizes; (void)n_in; (void)out_size; (void)ws_size;
  const float* x  = (const float*)d_in[0];
  const float* Wq = (const float*)d_in[1];
  const float* bq = (const float*)d_in[2];
  const float* Wk = (const float*)d_in[3];
  const float* bk = (const float*)d_in[4];
  const float* Wv = (const float*)d_in[5];
  const float* bv = (const float*)d_in[6];
  const float* Wo = (const float*)d_in[7];
  const float* bo = (const float*)d_in[8];
  const float* g  = (const float*)d_in[9];
  const float* be = (const float*)d_in[10];
  float* out = (float*)d_out;

  float* kv_acc = (float*)d_ws;               // 8*32*32 floats
  float* ksum_acc = kv_acc + H * HD * HD;     // 8*32 floats
  const int acc_n = H * HD * HD + H * HD;     // 8448

  const int lds1 = 3 * EB * C * (int)sizeof(float);  // 196608 B
  const int lds2 = (3 * EB * C + H * HD * HD + C + EB * H + 8 * EB + 2 * EB) *
                   (int)sizeof(float);               // 235008 B
  hipFuncSetAttribute((const void*)pass1_kv,
                      hipFuncAttributeMaxDynamicSharedMemorySize, lds1);
  hipFuncSetAttribute((const void*)pass2_out,
                      hipFuncAttributeMaxDynamicSharedMemorySize, lds2);

  zero_ws_kernel<<<(acc_n + 255) / 256, 256, 0, stream>>>(kv_acc, acc_n);
  pass1_kv<<<NB, THREADS, lds1, stream>>>(x, Wk, bk, Wv, bv, kv_acc, ksum_acc);
  pass2_out<<<NB, THREADS, lds2, stream>>>(x, Wq, bq, kv_acc, ksum_acc, Wo, bo,
                                           g, be, out);
}